// TransformerModel_63479616635163
// MI455X (gfx1250) — compile-verified
//
#include <hip/hip_runtime.h>
#include <math.h>

// Problem constants (fixed by the reference).
#define B_   4
#define M_   10
#define S_   1024
#define F_   128
#define E_   512
#define H_   8
#define L_   6
#define DH_  64
#define NVNP 18
#define ROWS (B_ * S_)   // 4096

typedef __attribute__((ext_vector_type(2))) float v2f;
typedef __attribute__((ext_vector_type(4))) float v4f;
typedef __attribute__((ext_vector_type(8))) float v8f;

__device__ __forceinline__ v8f zero8() {
  v8f z;
#pragma unroll
  for (int i = 0; i < 8; ++i) z[i] = 0.f;
  return z;
}

// D = A(16x4, f32) * B(4x16, f32) + C(16x16, f32)  -- full-precision CDNA5 WMMA.
// Fragment layout (ISA 7.12.2): A: lane holds M=lane%16, K = 2*(lane/16)+v.
//                               B: lane holds N=lane%16, K = 2*(lane/16)+v.
//                               C/D: VGPR r holds row r+8*(lane/16), col lane%16.
__device__ __forceinline__ v8f wmma4(v2f a, v2f b, v8f c) {
  return __builtin_amdgcn_wmma_f32_16x16x4_f32(false, a, false, b, (short)0, c,
                                               false, false);
}

// ---------------------------------------------------------------------------
// 1) Mean over ensemble members: favg[b,s,f] = mean_m features[b,m,s,f]
// ---------------------------------------------------------------------------
__global__ __launch_bounds__(256) void k_feat_mean(const float* __restrict__ f,
                                                   float* __restrict__ favg) {
  int idx = blockIdx.x * 256 + threadIdx.x;  // [0, B*S*F)
  int b   = idx / (S_ * F_);
  int rem = idx - b * (S_ * F_);
  const float* p = f + (size_t)b * M_ * S_ * F_ + rem;
  float s = 0.f;
#pragma unroll
  for (int m = 0; m < M_; ++m) s += p[(size_t)m * S_ * F_];
  favg[idx] = s * (1.0f / M_);
}

// ---------------------------------------------------------------------------
// 2) WMMA f32 GEMM: C[ROWS,N] = A[ROWS,K] @ W[K,N] (+bias) (+silu)
//    block = 256 thr (8 waves), tile 64x128; wave tile 16x64.
//    W tile staged TRANSPOSED in LDS so each B fragment is one ds_load_b64
//    into an aligned VGPR pair (no v_mov glue before v_wmma).
// ---------------------------------------------------------------------------
__global__ __launch_bounds__(256) void k_gemm(const float* __restrict__ A,
                                              const float* __restrict__ W,
                                              const float* __restrict__ bias,
                                              float* __restrict__ C, int K,
                                              int N, int act) {
  __shared__ __align__(16) float a_lds[64][32];
  __shared__ __align__(16) float w_t[128][34];  // transposed, even pad
  const int tid  = threadIdx.x;
  const int wave = tid >> 5, lane = tid & 31;
  const int wr = wave & 3;    // wave row   (4 x 16 rows)
  const int wc = wave >> 2;   // wave col   (2 x 64 cols)
  const int mrow = lane & 15, kh = lane >> 4;
  const int row0 = blockIdx.x * 64;
  const int col0 = blockIdx.y * 128;
  const int kq   = wave * 4;  // 4 k-rows per wave for W staging

  v8f acc[4];
#pragma unroll
  for (int i = 0; i < 4; ++i) acc[i] = zero8();

  for (int k0 = 0; k0 < K; k0 += 32) {
    // stage A (64x32) row-major, float4 cooperative loads
    for (int i = tid; i < 512; i += 256) {
      int ar = i >> 3, ac = (i & 7) << 2;
      *(v4f*)&a_lds[ar][ac] =
          *(const v4f*)&A[(size_t)(row0 + ar) * K + k0 + ac];
    }
    // stage W (32x128) transposed: lanes sweep consecutive columns (coalesced
    // global reads); LDS store lane-stride 34 dwords -> conflict-free banks.
#pragma unroll
    for (int cb = 0; cb < 4; ++cb) {
      int c = cb * 32 + lane;
#pragma unroll
      for (int kr = 0; kr < 4; ++kr)
        w_t[c][kq + kr] = W[(size_t)(k0 + kq + kr) * N + col0 + c];
    }
    // prefetch next K-chunk while this one is consumed (global_prefetch_b8)
    if (k0 + 32 < K) {
      __builtin_prefetch(&A[(size_t)(row0 + (tid >> 2)) * K + k0 + 32 +
                            (tid & 3) * 8], 0, 1);
      __builtin_prefetch(&W[(size_t)(k0 + 32 + kq) * N + col0 + lane * 4], 0, 1);
    }
    __syncthreads();
#pragma unroll
    for (int kk = 0; kk < 8; ++kk) {
      v2f af = *(const v2f*)&a_lds[wr * 16 + mrow][kk * 4 + 2 * kh];
#pragma unroll
      for (int nt = 0; nt < 4; ++nt) {
        v2f wf = *(const v2f*)&w_t[wc * 64 + nt * 16 + mrow][kk * 4 + 2 * kh];
        acc[nt] = wmma4(af, wf, acc[nt]);
      }
    }
    __syncthreads();
  }

#pragma unroll
  for (int nt = 0; nt < 4; ++nt) {
    int col  = col0 + wc * 64 + nt * 16 + mrow;
    float bv = bias ? bias[col] : 0.f;
#pragma unroll
    for (int r = 0; r < 8; ++r) {
      int row   = row0 + wr * 16 + r + 8 * kh;
      float val = acc[nt][r] + bv;
      if (act) val = val / (1.f + expf(-val));  // SiLU
      C[(size_t)row * N + col] = val;
    }
  }
}

// ---------------------------------------------------------------------------
// 3) Station-embedding add with NaN-target gating (sid -1 -> last row S_)
// ---------------------------------------------------------------------------
__global__ __launch_bounds__(256) void k_station_add(
    float* __restrict__ x, const float* __restrict__ se,
    const float* __restrict__ target) {
  int row = blockIdx.x;
  int b = row >> 10, s = row & (S_ - 1);
  const float* t = target + ((size_t)b * S_ + s) * 6;
  bool nan = false;
#pragma unroll
  for (int j = 0; j < 6; ++j) {
    float v = t[j];
    nan = nan || (v != v);
  }
  int sid = nan ? S_ : s;
  const float* sp = se + (size_t)sid * E_;
  float* xp = x + (size_t)row * E_;
  for (int c = threadIdx.x; c < E_; c += 256) xp[c] += sp[c];
}

// ---------------------------------------------------------------------------
// 4) Attention with batch-axis softmax (softmax over B=4 per (h,q,k)).
//    1 block per (head, 16-row q tile); 4 waves = 4 batches.
//    Streaming: no [H,B,S,S] intermediate.
// ---------------------------------------------------------------------------
__global__ __launch_bounds__(128) void k_attn(const float* __restrict__ q,
                                              const float* __restrict__ k,
                                              const float* __restrict__ v,
                                              float* __restrict__ o) {
  __shared__ float s_lds[B_][16][17];
  __shared__ __align__(16) float p_lds[B_][16][18];  // even pad: b64 readback
  const int lane = threadIdx.x & 31;
  const int b    = threadIdx.x >> 5;  // wave == batch
  const int mrow = lane & 15, kh = lane >> 4;
  const int h  = blockIdx.x >> 6;
  const int qt = blockIdx.x & 63;

  // Q A-fragments kept in registers for the whole K sweep (16 x v2f).
  const float* qp = q + (size_t)(b * S_ + qt * 16 + mrow) * E_ + h * DH_;
  v2f qf[16];
#pragma unroll
  for (int kk = 0; kk < 16; ++kk) qf[kk] = *(const v2f*)(qp + kk * 4 + 2 * kh);

  v8f acc[4];
#pragma unroll
  for (int i = 0; i < 4; ++i) acc[i] = zero8();

  for (int kt = 0; kt < 64; ++kt) {
    // scores: S = Q(16x64) * K^T(64x16) / 8     (16 WMMAs, K streamed from L2)
    const float* kp = k + (size_t)(b * S_ + kt * 16 + mrow) * E_ + h * DH_;
    v8f sc = zero8();
#pragma unroll
    for (int kk = 0; kk < 16; ++kk) {
      v2f kf = *(const v2f*)(kp + kk * 4 + 2 * kh);
      sc = wmma4(qf[kk], kf, sc);
    }
#pragma unroll
    for (int r = 0; r < 8; ++r)
      s_lds[b][r + 8 * kh][mrow] = sc[r] * 0.125f;  // 1/sqrt(DH)
    __syncthreads();

    // elementwise softmax across the 4 batches
    float p[8];
#pragma unroll
    for (int r = 0; r < 8; ++r) {
      int rr   = r + 8 * kh;
      float s0 = s_lds[0][rr][mrow], s1 = s_lds[1][rr][mrow];
      float s2 = s_lds[2][rr][mrow], s3 = s_lds[3][rr][mrow];
      float mx = fmaxf(fmaxf(s0, s1), fmaxf(s2, s3));
      float e0 = expf(s0 - mx), e1 = expf(s1 - mx);
      float e2 = expf(s2 - mx), e3 = expf(s3 - mx);
      float inv = 1.f / (e0 + e1 + e2 + e3);
      float eb  = (b == 0) ? e0 : ((b == 1) ? e1 : ((b == 2) ? e2 : e3));
      p[r] = eb * inv;
    }
    __syncthreads();  // all s reads done before next kt overwrites s_lds

    // re-fragment P (C-layout -> A-layout) via own LDS slot
#pragma unroll
    for (int r = 0; r < 8; ++r) p_lds[b][r + 8 * kh][mrow] = p[r];
    v2f pf[4];
#pragma unroll
    for (int kk2 = 0; kk2 < 4; ++kk2)
      pf[kk2] = *(const v2f*)&p_lds[b][mrow][kk2 * 4 + 2 * kh];

    // O += P(16x16) * V(16x64)                  (16 WMMAs)
    const float* vp = v + (size_t)(b * S_ + kt * 16) * E_ + h * DH_;
#pragma unroll
    for (int nt = 0; nt < 4; ++nt) {
#pragma unroll
      for (int kk2 = 0; kk2 < 4; ++kk2) {
        int kr = kk2 * 4 + 2 * kh;
        v2f vf;
        vf.x = vp[(size_t)kr * E_ + nt * 16 + mrow];
        vf.y = vp[(size_t)(kr + 1) * E_ + nt * 16 + mrow];
        acc[nt] = wmma4(pf[kk2], vf, acc[nt]);
      }
    }
  }

#pragma unroll
  for (int nt = 0; nt < 4; ++nt)
#pragma unroll
    for (int r = 0; r < 8; ++r)
      o[(size_t)(b * S_ + qt * 16 + r + 8 * kh) * E_ + h * DH_ + nt * 16 +
        mrow] = acc[nt][r];
}

// ---------------------------------------------------------------------------
// 5) Fused row kernels: LN->SiLU->LN and (a+b)->LN. 256 thr, 2 elems/thr.
// ---------------------------------------------------------------------------
__device__ __forceinline__ void block_red2(float& a, float& b, float* sm) {
#pragma unroll
  for (int off = 16; off > 0; off >>= 1) {
    a += __shfl_xor(a, off, 32);
    b += __shfl_xor(b, off, 32);
  }
  int wave = threadIdx.x >> 5, lane = threadIdx.x & 31;
  if (lane == 0) {
    sm[wave * 2] = a;
    sm[wave * 2 + 1] = b;
  }
  __syncthreads();
  float sa = 0.f, sb = 0.f;
#pragma unroll
  for (int i = 0; i < 8; ++i) {
    sa += sm[2 * i];
    sb += sm[2 * i + 1];
  }
  __syncthreads();
  a = sa;
  b = sb;
}

__global__ __launch_bounds__(256) void k_ln_silu_ln(
    const float* __restrict__ in, const float* __restrict__ g1,
    const float* __restrict__ bb1, const float* __restrict__ g2,
    const float* __restrict__ bb2, float* __restrict__ out) {
  __shared__ float sm[16];
  int row = blockIdx.x, tid = threadIdx.x;
  const float* rp = in + (size_t)row * E_;
  float v0 = rp[tid], v1 = rp[tid + 256];
  float s = v0 + v1, q = v0 * v0 + v1 * v1;
  block_red2(s, q, sm);
  float mu = s * (1.f / E_);
  float rs = rsqrtf(fmaxf(q * (1.f / E_) - mu * mu, 0.f) + 1e-5f);
  float u0 = (v0 - mu) * rs * g1[tid] + bb1[tid];
  float u1 = (v1 - mu) * rs * g1[tid + 256] + bb1[tid + 256];
  float t0 = u0 / (1.f + expf(-u0));
  float t1 = u1 / (1.f + expf(-u1));
  s = t0 + t1;
  q = t0 * t0 + t1 * t1;
  block_red2(s, q, sm);
  mu = s * (1.f / E_);
  rs = rsqrtf(fmaxf(q * (1.f / E_) - mu * mu, 0.f) + 1e-5f);
  out[(size_t)row * E_ + tid]       = (t0 - mu) * rs * g2[tid] + bb2[tid];
  out[(size_t)row * E_ + tid + 256] =
      (t1 - mu) * rs * g2[tid + 256] + bb2[tid + 256];
}

__global__ __launch_bounds__(256) void k_add_ln(const float* __restrict__ a,
                                                const float* __restrict__ b,
                                                const float* __restrict__ g,
                                                const float* __restrict__ bb,
                                                float* __restrict__ out) {
  __shared__ float sm[16];
  int row = blockIdx.x, tid = threadIdx.x;
  float v0 = a[(size_t)row * E_ + tid] + b[(size_t)row * E_ + tid];
  float v1 = a[(size_t)row * E_ + tid + 256] + b[(size_t)row * E_ + tid + 256];
  float s = v0 + v1, q = v0 * v0 + v1 * v1;
  block_red2(s, q, sm);
  float mu = s * (1.f / E_);
  float rs = rsqrtf(fmaxf(q * (1.f / E_) - mu * mu, 0.f) + 1e-5f);
  out[(size_t)row * E_ + tid]       = (v0 - mu) * rs * g[tid] + bb[tid];
  out[(size_t)row * E_ + tid + 256] =
      (v1 - mu) * rs * g[tid + 256] + bb[tid + 256];
}

// ---------------------------------------------------------------------------
// 6) Regression head: out[row, j<18] = x[row,:] @ reg_W[:,j] + reg_b[j]
// ---------------------------------------------------------------------------
__global__ __launch_bounds__(64) void k_head(const float* __restrict__ x,
                                             const float* __restrict__ W,
                                             const float* __restrict__ bias,
                                             float* __restrict__ out) {
  __shared__ float xr[E_];
  int row = blockIdx.x;
  for (int c = threadIdx.x; c < E_; c += 64) xr[c] = x[(size_t)row * E_ + c];
  __syncthreads();
  int j = threadIdx.x;
  if (j < NVNP) {
    float s = bias[j];
    for (int kk = 0; kk < E_; ++kk) s += xr[kk] * W[kk * NVNP + j];
    out[(size_t)row * NVNP + j] = s;
  }
}

// ---------------------------------------------------------------------------
// Orchestration. Workspace (floats): favg(0.5M) x(2M) q(2M) k(2M) v(2M) o(2M)
// = 44 MB total. FFN temps alias q/k/v.
// ---------------------------------------------------------------------------
extern "C" void kernel_launch(void* const* d_in, const int* in_sizes, int n_in,
                              void* d_out, int out_size, void* d_ws,
                              size_t ws_size, hipStream_t stream) {
  const float* features = (const float*)d_in[0];
  const float* target   = (const float*)d_in[1];
  // d_in[2] step_idx, d_in[3] day_of_year: unused by the reference math.
  const float* emb_W  = (const float*)d_in[4];
  const float* emb_b  = (const float*)d_in[5];
  const float* st_emb = (const float*)d_in[6];
  const float* Wq  = (const float*)d_in[7];
  const float* Wk  = (const float*)d_in[8];
  const float* Wv  = (const float*)d_in[9];
  const float* aW  = (const float*)d_in[10];
  const float* ab  = (const float*)d_in[11];
  const float* lag = (const float*)d_in[12];
  const float* lab = (const float*)d_in[13];
  const float* W1  = (const float*)d_in[14];
  const float* b1  = (const float*)d_in[15];
  const float* W2  = (const float*)d_in[16];
  const float* b2  = (const float*)d_in[17];
  const float* l1g = (const float*)d_in[18];
  const float* l1b = (const float*)d_in[19];
  const float* l2g = (const float*)d_in[20];
  const float* l2b = (const float*)d_in[21];
  const float* regW = (const float*)d_in[22];
  const float* regb = (const float*)d_in[23];
  float* out = (float*)d_out;

  const size_t SZ_X = (size_t)ROWS * E_;
  const size_t EE   = (size_t)E_ * E_;
  float* ws   = (float*)d_ws;
  float* favg = ws;
  float* x  = favg + (size_t)B_ * S_ * F_;
  float* qb = x + SZ_X;
  float* kb = qb + SZ_X;
  float* vb = kb + SZ_X;
  float* ob = vb + SZ_X;

  dim3 gg(ROWS / 64, E_ / 128);

  k_feat_mean<<<(B_ * S_ * F_) / 256, 256, 0, stream>>>(features, favg);
  k_gemm<<<gg, 256, 0, stream>>>(favg, emb_W, emb_b, x, F_, E_, 0);
  k_station_add<<<ROWS, 256, 0, stream>>>(x, st_emb, target);

  for (int l = 0; l < L_; ++l) {
    k_gemm<<<gg, 256, 0, stream>>>(x, Wq + l * EE, nullptr, qb, E_, E_, 0);
    k_gemm<<<gg, 256, 0, stream>>>(x, Wk + l * EE, nullptr, kb, E_, E_, 0);
    k_gemm<<<gg, 256, 0, stream>>>(x, Wv + l * EE, nullptr, vb, E_, E_, 0);
    k_attn<<<H_ * (S_ / 16), 128, 0, stream>>>(qb, kb, vb, ob);
    k_gemm<<<gg, 256, 0, stream>>>(ob, aW + l * EE, ab + l * E_, qb, E_, E_, 0);
    k_ln_silu_ln<<<ROWS, 256, 0, stream>>>(qb, lag + l * E_, lab + l * E_,
                                           l1g + l * E_, l1b + l * E_, kb);
    k_gemm<<<gg, 256, 0, stream>>>(kb, W1 + l * EE, b1 + l * E_, vb, E_, E_, 1);
    k_gemm<<<gg, 256, 0, stream>>>(vb, W2 + l * EE, b2 + l * E_, qb, E_, E_, 0);
    k_add_ln<<<ROWS, 256, 0, stream>>>(qb, kb, l2g + l * E_, l2b + l * E_, x);
  }

  k_head<<<ROWS, 64, 0, stream>>>(x, regW, regb, out);
}